// AttentionPITF_1211180777753
// MI455X (gfx1250) — compile-verified
//
#include <hip/hip_runtime.h>

typedef __attribute__((ext_vector_type(2))) float v2f;
typedef __attribute__((ext_vector_type(8))) float v8f;

#define KDIM   64
#define MHIST  50
#define XCOLS  54          // 4 + M
#define WPB    8           // waves per block (wave32)
#define GAMMA  0.5f

// Dynamic LDS layout (float units):
//   [0, 4096)                      W      (64x64, row-major)
//   [4096, 4160)                   bias   (64)
//   [4160 + w*4096, ... )          hist   per-wave 64x64 tile (rows 50..63 zero)
//   [4160 + WPB*4096 + w*64, ...)  logits/alpha scratch per wave
#define SMEM_FLOATS (4160 + WPB * 4096 + WPB * 64)

__global__ __launch_bounds__(WPB * 32) void attn_pitf_kernel(
    const int* __restrict__ x,
    const float* __restrict__ userVecs,
    const float* __restrict__ itemVecs,
    const float* __restrict__ tagUserVecs,
    const float* __restrict__ tagItemVecs,
    const float* __restrict__ W,
    const float* __restrict__ bias,
    float* __restrict__ out,
    int nBatch)
{
    extern __shared__ float smem[];
    float* W_sh    = smem;
    float* bias_sh = smem + 4096;

    const int tid  = threadIdx.x;
    const int wave = tid >> 5;
    const int lane = tid & 31;

    float* hist_sh = smem + 4160 + wave * 4096;
    float* scr_sh  = smem + 4160 + WPB * 4096 + wave * 64;

    // ---- Stage W (64x64) and bias into LDS, shared by all 8 waves ----
    {
        const float4* Wg = (const float4*)W;
        float4*       Ws = (float4*)W_sh;
#pragma unroll
        for (int q = 0; q < 4; ++q) Ws[tid + q * 256] = Wg[tid + q * 256];
        if (tid < 16) ((float4*)bias_sh)[tid] = ((const float4*)bias)[tid];
    }
    __syncthreads();

    const int b = blockIdx.x * WPB + wave;
    if (b >= nBatch) return;

    const int* xb = x + (long long)b * XCOLS;
    const int x0 = xb[0], x1 = xb[1], x2 = xb[2], x3 = xb[3];

    const int jcol = lane & 15;   // column within 16-wide fragment
    const int half = lane >> 4;   // half-wave selects K-pair / row parity

    // ---- Gather 50 hist rows (tagUserVecs[x[b,4..53]]) into LDS; pad to 64 rows ----
    {
        const int q = jcol;                  // float4 slot within a row (16 per row)
#pragma unroll 1
        for (int rr = 0; rr < 25; ++rr) {    // 2 rows per iteration
            const int r = rr * 2 + half;
            const long long tag = (long long)xb[4 + r];
            float4 v = ((const float4*)(tagUserVecs + tag * KDIM))[q];
            ((float4*)(hist_sh + r * KDIM))[q] = v;
        }
        const float4 z = make_float4(0.f, 0.f, 0.f, 0.f);
#pragma unroll
        for (int zr = 0; zr < 7; ++zr) {     // zero rows 50..63
            const int r = 50 + zr * 2 + half;
            ((float4*)(hist_sh + r * KDIM))[q] = z;
        }
    }
    asm volatile("s_wait_dscnt 0" ::: "memory");

    // Per-lane bias and u values for the fused epilogue (column = n*16 + jcol)
    float bvals[4], uvals[4];
#pragma unroll
    for (int n = 0; n < 4; ++n) {
        bvals[n] = bias_sh[n * 16 + jcol];
        uvals[n] = userVecs[(long long)x0 * KDIM + n * 16 + jcol];
    }

    // ---- WMMA GEMM: tag_h = relu(hist * W^T + bias), fused dot with u -> logits ----
    // A (16x4 f32): lane = row (both halves), VGPRs = K pair; half selects K {0,1} vs {2,3}
    // B (4x16 f32): lane = output column j (W row j), half selects K pair. B[k][j] = W[j][k].
#pragma unroll 1
    for (int mt = 0; mt < 4; ++mt) {
        const int row = mt * 16 + jcol;
        v2f a[16];
#pragma unroll
        for (int s = 0; s < 16; ++s)
            a[s] = *(const v2f*)(hist_sh + row * KDIM + s * 4 + half * 2);

        v8f c0 = {}, c1 = {}, c2 = {}, c3 = {};
#pragma unroll
        for (int s = 0; s < 16; ++s) {
            const int ko = s * 4 + half * 2;
            v2f b0 = *(const v2f*)(W_sh + (0 * 16 + jcol) * KDIM + ko);
            v2f b1 = *(const v2f*)(W_sh + (1 * 16 + jcol) * KDIM + ko);
            v2f b2 = *(const v2f*)(W_sh + (2 * 16 + jcol) * KDIM + ko);
            v2f b3 = *(const v2f*)(W_sh + (3 * 16 + jcol) * KDIM + ko);
            c0 = __builtin_amdgcn_wmma_f32_16x16x4_f32(false, a[s], false, b0, (short)0, c0, false, false);
            c1 = __builtin_amdgcn_wmma_f32_16x16x4_f32(false, a[s], false, b1, (short)0, c1, false, false);
            c2 = __builtin_amdgcn_wmma_f32_16x16x4_f32(false, a[s], false, b2, (short)0, c2, false, false);
            c3 = __builtin_amdgcn_wmma_f32_16x16x4_f32(false, a[s], false, b3, (short)0, c3, false, false);
        }

        // Fused epilogue: logits row = sum_j relu(tag_h + bias[j]) * u[j]
        // C/D layout: VGPR v, lane l -> M = v + 8*half, N = jcol
#pragma unroll
        for (int v = 0; v < 8; ++v) {
            float p = fmaxf(c0[v] + bvals[0], 0.f) * uvals[0]
                    + fmaxf(c1[v] + bvals[1], 0.f) * uvals[1]
                    + fmaxf(c2[v] + bvals[2], 0.f) * uvals[2]
                    + fmaxf(c3[v] + bvals[3], 0.f) * uvals[3];
            p += __shfl_xor(p, 1);
            p += __shfl_xor(p, 2);
            p += __shfl_xor(p, 4);
            p += __shfl_xor(p, 8);
            if (jcol == 0) scr_sh[mt * 16 + half * 8 + v] = p;
        }
    }
    asm volatile("s_wait_dscnt 0" ::: "memory");

    // ---- Softmax over m (50 valid rows), lane handles m0 = lane, m1 = lane+32 ----
    const int m0 = lane, m1 = lane + 32;
    float v0 = (m0 < MHIST) ? scr_sh[m0] : -1e30f;
    float v1 = (m1 < MHIST) ? scr_sh[m1] : -1e30f;
    float mx = fmaxf(v0, v1);
#pragma unroll
    for (int off = 1; off < 32; off <<= 1) mx = fmaxf(mx, __shfl_xor(mx, off));
    float e0 = (m0 < MHIST) ? __expf(v0 - mx) : 0.f;
    float e1 = (m1 < MHIST) ? __expf(v1 - mx) : 0.f;
    float ssum = e0 + e1;
#pragma unroll
    for (int off = 1; off < 32; off <<= 1) ssum += __shfl_xor(ssum, off);
    const float inv = 1.f / ssum;
    scr_sh[m0] = e0 * inv;
    scr_sh[m1] = e1 * inv;
    asm volatile("s_wait_dscnt 0" ::: "memory");

    // ---- h = sum_m alpha[m] * hist[m], lane handles k0 = lane, k1 = lane+32 ----
    const int k0 = lane, k1 = lane + 32;
    float h0 = 0.f, h1 = 0.f;
#pragma unroll 2
    for (int m = 0; m < MHIST; ++m) {
        const float am = scr_sh[m];
        h0 = fmaf(am, hist_sh[m * KDIM + k0], h0);
        h1 = fmaf(am, hist_sh[m * KDIM + k1], h1);
    }

    // ---- mix = (1-g)u + g*h;  r = sum_k mix*(tu-ntu) + i*(ti-nti) ----
    const float u0 = userVecs[(long long)x0 * KDIM + k0];
    const float u1 = userVecs[(long long)x0 * KDIM + k1];
    const float i0 = itemVecs[(long long)x1 * KDIM + k0];
    const float i1 = itemVecs[(long long)x1 * KDIM + k1];
    const float tu0 = tagUserVecs[(long long)x2 * KDIM + k0];
    const float tu1 = tagUserVecs[(long long)x2 * KDIM + k1];
    const float ti0 = tagItemVecs[(long long)x2 * KDIM + k0];
    const float ti1 = tagItemVecs[(long long)x2 * KDIM + k1];
    const float ntu0 = tagUserVecs[(long long)x3 * KDIM + k0];
    const float ntu1 = tagUserVecs[(long long)x3 * KDIM + k1];
    const float nti0 = tagItemVecs[(long long)x3 * KDIM + k0];
    const float nti1 = tagItemVecs[(long long)x3 * KDIM + k1];

    const float mix0 = (1.f - GAMMA) * u0 + GAMMA * h0;
    const float mix1 = (1.f - GAMMA) * u1 + GAMMA * h1;
    float r = mix0 * (tu0 - ntu0) + i0 * (ti0 - nti0)
            + mix1 * (tu1 - ntu1) + i1 * (ti1 - nti1);
#pragma unroll
    for (int off = 1; off < 32; off <<= 1) r += __shfl_xor(r, off);
    if (lane == 0) out[b] = r;
}

extern "C" void kernel_launch(void* const* d_in, const int* in_sizes, int n_in,
                              void* d_out, int out_size, void* d_ws, size_t ws_size,
                              hipStream_t stream) {
    const int*   x           = (const int*)d_in[0];
    const float* userVecs    = (const float*)d_in[1];
    const float* itemVecs    = (const float*)d_in[2];
    const float* tagUserVecs = (const float*)d_in[3];
    const float* tagItemVecs = (const float*)d_in[4];
    const float* W           = (const float*)d_in[5];
    const float* bias        = (const float*)d_in[6];
    float*       out         = (float*)d_out;

    const int nBatch = in_sizes[0] / XCOLS;      // 16384
    const dim3 block(WPB * 32);
    const dim3 grid((nBatch + WPB - 1) / WPB);
    const size_t shmem = (size_t)SMEM_FLOATS * sizeof(float);

    attn_pitf_kernel<<<grid, block, shmem, stream>>>(
        x, userVecs, itemVecs, tagUserVecs, tagItemVecs, W, bias, out, nBatch);
}